// PointNetSetAbstraction_79070347920081
// MI455X (gfx1250) — compile-verified
//
#include <hip/hip_runtime.h>

#define B_   32
#define C_   64
#define N_   1024
#define K_   10
#define OUT_ 1024
#define KC_  (K_ * C_)   // 640
#define EPSV 1e-5f

typedef float v2f __attribute__((ext_vector_type(2)));
typedef float v8f __attribute__((ext_vector_type(8)));

// ---------------------------------------------------------------------------
// Stage 1: per-batch kNN top-10 (neg squared distance), batch resident in LDS.
// Grid: (4, B), block 256. LDS: 64x1024 f32 = 256KB (+4KB norms) of 320KB WGP LDS.
// ---------------------------------------------------------------------------
__launch_bounds__(256)
__global__ void topk_kernel(const float* __restrict__ x, int* __restrict__ idx_out) {
    __shared__ float xs[C_ * N_];   // 256 KB
    __shared__ float xx[N_];        // 4 KB
    const int b   = blockIdx.y;
    const int tid = threadIdx.x;

    // cooperative load of x[b] (contiguous 64*1024 f32) into LDS as float4
    const float4* xg4 = (const float4*)(x + (size_t)b * C_ * N_);
    float4* xs4 = (float4*)xs;
    for (int i = tid; i < (C_ * N_) / 4; i += 256) xs4[i] = xg4[i];
    __syncthreads();

    // per-point squared norms
    for (int m = tid; m < N_; m += 256) {
        float s = 0.f;
        #pragma unroll
        for (int c = 0; c < C_; ++c) { float v = xs[c * N_ + m]; s = fmaf(v, v, s); }
        xx[m] = s;
    }
    __syncthreads();

    const int n = blockIdx.x * 256 + tid;   // this thread's point
    float xn[C_];
    #pragma unroll
    for (int c = 0; c < C_; ++c) xn[c] = xs[c * N_ + n];
    const float xxn = xx[n];

    float dk[K_]; int ik[K_];
    #pragma unroll
    for (int j = 0; j < K_; ++j) { dk[j] = -3.4e38f; ik[j] = 0; }

    for (int m = 0; m < N_; ++m) {
        float acc = 0.f;
        #pragma unroll
        for (int c = 0; c < C_; ++c) acc = fmaf(xn[c], xs[c * N_ + m], acc); // LDS broadcast
        float cv = 2.f * acc - xxn - xx[m];   // neg squared distance
        int   ci = m;
        // branch-free sorted insertion (descending); strict > keeps lower index on ties
        #pragma unroll
        for (int j = 0; j < K_; ++j) {
            bool  gt = cv > dk[j];
            float td = gt ? dk[j] : cv;
            int   ti = gt ? ik[j] : ci;
            dk[j] = gt ? cv : dk[j];
            ik[j] = gt ? ci : ik[j];
            cv = td; ci = ti;
        }
    }
    int* o = idx_out + ((size_t)b * N_ + n) * K_;
    #pragma unroll
    for (int j = 0; j < K_; ++j) o[j] = ik[j];
}

// ---------------------------------------------------------------------------
// Stage 2/4: fp32 GEMM with V_WMMA_F32_16X16X4_F32.
//   Cout[b] (1024 x 640) = A (1024 x 1024, row-major) * B[b] (1024 x 640)
//   GATHER=true : B[n][j] = x[b, j&63, idx[b, n, j>>6]]   (fused kNN gather)
//   GATHER=false: B = Bmat[b] dense, ld = 640
// Block: 256 thr = 8 waves (4 x 2); block tile 128(M) x 64(N); wave tile 32x32
// (2x2 accumulators -> 4 WMMAs per 4 LDS b64 operand loads). K-chunk 32.
// LDS tiles padded to even stride (34 f32) so each WMMA operand is one
// 8-byte-aligned contiguous 64-bit LDS read; B tile stored transposed.
// ---------------------------------------------------------------------------
template <bool GATHER>
__launch_bounds__(256)
__global__ void gemm_wmma_f32(const float* __restrict__ A,
                              const float* __restrict__ Bmat,
                              const int*   __restrict__ nbr,
                              float* __restrict__ Cout) {
    constexpr int KD  = 1024;
    constexpr int KCH = 32;
    constexpr int LDA = KCH + 2;   // 34: even => b64-aligned, 17*i mod 32 perm => no conflicts
    __shared__ __align__(16) float As [128][LDA];  // [row][k]
    __shared__ __align__(16) float Bst[64][LDA];   // [col][k] (transposed)

    const int b    = blockIdx.z;
    const int m0   = blockIdx.x * 128;
    const int n0   = blockIdx.y * 64;
    const int tid  = threadIdx.x;
    const int lane = tid & 31;
    const int wave = tid >> 5;
    const int wm   = wave & 3;    // 0..3 along M (32 rows each)
    const int wn   = wave >> 2;   // 0..1 along N (32 cols each)

    const float* Bb   = GATHER ? Bmat + (size_t)b * C_ * N_
                               : Bmat + (size_t)b * KD * KC_;
    const int*   idxb = GATHER ? nbr + (size_t)b * N_ * K_ : nullptr;

    v8f acc00 = {0.f,0.f,0.f,0.f,0.f,0.f,0.f,0.f};
    v8f acc01 = acc00, acc10 = acc00, acc11 = acc00;

    const int hrow  = lane & 15;           // row within 16 (A) / col within 16 (B)
    const int khalf = (lane >> 4) * 2;     // lanes 0-15: K=0,1 ; lanes 16-31: K=2,3
    const int ra0   = wm * 32 + hrow;      // A rows for this wave
    const int ra1   = ra0 + 16;
    const int cb0   = wn * 32 + hrow;      // B cols for this wave
    const int cb1   = cb0 + 16;

    for (int kc = 0; kc < KD; kc += KCH) {
        // ---- stage A tile: 128 x 32 (float4, coalesced) ----
        for (int i = tid; i < 128 * 8; i += 256) {
            int r = i >> 3, k4 = (i & 7) * 4;
            float4 v = *(const float4*)&A[(size_t)(m0 + r) * KD + kc + k4];
            As[r][k4] = v.x; As[r][k4 + 1] = v.y; As[r][k4 + 2] = v.z; As[r][k4 + 3] = v.w;
        }
        // prefetch next A K-chunk (global_prefetch_b8)
        if (kc + KCH < KD && tid < 128)
            __builtin_prefetch(&A[(size_t)(m0 + tid) * KD + kc + KCH], 0, 1);
        // ---- stage B tile (transposed): Bst[col][k] ----
        if (GATHER) {
            for (int i = tid; i < KCH * 64; i += 256) {
                int kk = i >> 6, jj = i & 63;            // conflict-free: stride-34 writes
                int j   = n0 + jj;                       // output col -> (k = j>>6, c = j&63)
                int src = idxb[(size_t)(kc + kk) * K_ + (j >> 6)];
                Bst[jj][kk] = Bb[(size_t)(j & 63) * N_ + src];
            }
        } else {
            for (int i = tid; i < KCH * 16; i += 256) {
                int kk = i >> 4, j4 = (i & 15) * 4;
                float4 v = *(const float4*)&Bb[(size_t)(kc + kk) * KC_ + n0 + j4];
                Bst[j4][kk] = v.x; Bst[j4 + 1][kk] = v.y;
                Bst[j4 + 2][kk] = v.z; Bst[j4 + 3][kk] = v.w;
            }
        }
        __syncthreads();

        // ---- 8 K-steps x 4 WMMAs (2x2 accumulator grid) ----
        #pragma unroll
        for (int ks = 0; ks < KCH; ks += 4) {
            const v2f a0 = *(const v2f*)&As [ra0][ks + khalf];
            const v2f a1 = *(const v2f*)&As [ra1][ks + khalf];
            const v2f b0 = *(const v2f*)&Bst[cb0][ks + khalf];
            const v2f b1 = *(const v2f*)&Bst[cb1][ks + khalf];
            acc00 = __builtin_amdgcn_wmma_f32_16x16x4_f32(false, a0, false, b0,
                                                          (short)0, acc00, false, false);
            acc01 = __builtin_amdgcn_wmma_f32_16x16x4_f32(false, a0, false, b1,
                                                          (short)0, acc01, false, false);
            acc10 = __builtin_amdgcn_wmma_f32_16x16x4_f32(false, a1, false, b0,
                                                          (short)0, acc10, false, false);
            acc11 = __builtin_amdgcn_wmma_f32_16x16x4_f32(false, a1, false, b1,
                                                          (short)0, acc11, false, false);
        }
        __syncthreads();
    }

    // ---- store C: VGPR v -> row v + 8*(lane>>4), col lane&15 (per 16x16 tile) ----
    float* Cb  = Cout + (size_t)b * OUT_ * KC_;
    int mb0    = m0 + wm * 32 + (lane >> 4) * 8;
    int nc0    = n0 + wn * 32 + (lane & 15);
    #pragma unroll
    for (int v = 0; v < 8; ++v) {
        Cb[(size_t)(mb0 + v)      * KC_ + nc0]      = acc00[v];
        Cb[(size_t)(mb0 + v)      * KC_ + nc0 + 16] = acc01[v];
        Cb[(size_t)(mb0 + v + 16) * KC_ + nc0]      = acc10[v];
        Cb[(size_t)(mb0 + v + 16) * KC_ + nc0 + 16] = acc11[v];
    }
}

// ---------------------------------------------------------------------------
// BatchNorm statistics: one block per channel o, reduce over (b, k, c).
// (bias b1/b2 omitted: a per-channel constant cancels exactly in BN.)
// ---------------------------------------------------------------------------
__launch_bounds__(256)
__global__ void bn_stats(const float* __restrict__ y,
                         float* __restrict__ mean, float* __restrict__ rvar) {
    __shared__ float ssum[256], ssq[256];
    const int o = blockIdx.x, tid = threadIdx.x;
    float s = 0.f, q = 0.f;
    for (int i = tid; i < B_ * KC_; i += 256) {
        int bb = i / KC_, j = i - bb * KC_;
        float v = y[((size_t)bb * OUT_ + o) * KC_ + j];
        s += v; q = fmaf(v, v, q);
    }
    ssum[tid] = s; ssq[tid] = q;
    __syncthreads();
    for (int st = 128; st > 0; st >>= 1) {
        if (tid < st) { ssum[tid] += ssum[tid + st]; ssq[tid] += ssq[tid + st]; }
        __syncthreads();
    }
    if (tid == 0) {
        const float inv = 1.f / (float)(B_ * KC_);
        float mn  = ssum[0] * inv;
        float var = ssq[0] * inv - mn * mn;
        mean[o] = mn;
        rvar[o] = rsqrtf(var + EPSV);
    }
}

// normalize + scale + ReLU, in place
__launch_bounds__(256)
__global__ void bn_relu(float* __restrict__ y, const float* __restrict__ mean,
                        const float* __restrict__ rvar, const float* __restrict__ gamma,
                        const float* __restrict__ beta) {
    size_t i = (size_t)blockIdx.x * 256 + threadIdx.x;
    int o = (int)((i / KC_) % OUT_);
    float v = (y[i] - mean[o]) * rvar[o] * gamma[o] + beta[o];
    y[i] = v > 0.f ? v : 0.f;
}

// final: normalize + scale + max over k + ReLU (ReLU commutes with max)
__launch_bounds__(256)
__global__ void bn_relu_maxpool(const float* __restrict__ y2, const float* __restrict__ mean,
                                const float* __restrict__ rvar, const float* __restrict__ gamma,
                                const float* __restrict__ beta, float* __restrict__ out) {
    int i = blockIdx.x * 256 + threadIdx.x;       // over B*OUT*C
    int c = i & 63;
    int o = (i >> 6) & (OUT_ - 1);
    int bb = i >> 16;
    const float* p = y2 + ((size_t)bb * OUT_ + o) * KC_ + c;
    float g  = gamma[o] * rvar[o];
    float sh = beta[o] - mean[o] * g;
    float m = -3.4e38f;
    #pragma unroll
    for (int k = 0; k < K_; ++k) m = fmaxf(m, p[k * C_] * g + sh);
    out[i] = fmaxf(m, 0.f);
}

// ---------------------------------------------------------------------------
extern "C" void kernel_launch(void* const* d_in, const int* in_sizes, int n_in,
                              void* d_out, int out_size, void* d_ws, size_t ws_size,
                              hipStream_t stream) {
    const float* x  = (const float*)d_in[0];
    const float* w1 = (const float*)d_in[1];
    const float* g1 = (const float*)d_in[3];
    const float* be1 = (const float*)d_in[4];
    const float* w2 = (const float*)d_in[5];
    const float* g2 = (const float*)d_in[7];
    const float* be2 = (const float*)d_in[8];

    char* ws = (char*)d_ws;
    size_t off = 0;
    int* idx = (int*)(ws + off);      off += (size_t)B_ * N_ * K_ * sizeof(int);
    off = (off + 255) & ~(size_t)255;
    float* y1 = (float*)(ws + off);   off += (size_t)B_ * OUT_ * KC_ * sizeof(float);
    float* y2 = (float*)(ws + off);   off += (size_t)B_ * OUT_ * KC_ * sizeof(float);
    float* mean1 = (float*)(ws + off); off += OUT_ * sizeof(float);
    float* rvar1 = (float*)(ws + off); off += OUT_ * sizeof(float);
    float* mean2 = (float*)(ws + off); off += OUT_ * sizeof(float);
    float* rvar2 = (float*)(ws + off); off += OUT_ * sizeof(float);

    // 1) kNN top-10 per point
    topk_kernel<<<dim3(4, B_), 256, 0, stream>>>(x, idx);
    // 2) GEMM1 with fused gather: y1[b] = W1 * G[b]
    gemm_wmma_f32<true><<<dim3(OUT_ / 128, KC_ / 64, B_), 256, 0, stream>>>(w1, x, idx, y1);
    // 3) BN1 + ReLU
    bn_stats<<<OUT_, 256, 0, stream>>>(y1, mean1, rvar1);
    bn_relu<<<(B_ * OUT_ * KC_) / 256, 256, 0, stream>>>(y1, mean1, rvar1, g1, be1);
    // 4) GEMM2: y2[b] = W2 * y1n[b]
    gemm_wmma_f32<false><<<dim3(OUT_ / 128, KC_ / 64, B_), 256, 0, stream>>>(w2, y1, nullptr, y2);
    // 5) BN2 + ReLU + max over k
    bn_stats<<<OUT_, 256, 0, stream>>>(y2, mean2, rvar2);
    bn_relu_maxpool<<<(B_ * OUT_ * C_) / 256, 256, 0, stream>>>(y2, mean2, rvar2, g2, be2,
                                                                (float*)d_out);
    (void)in_sizes; (void)n_in; (void)out_size; (void)ws_size;
}